// SolidAngleOccp_3985729651033
// MI455X (gfx1250) — compile-verified
//
#include <hip/hip_runtime.h>
#include <hip/hip_bf16.h>
#include <math.h>

typedef __attribute__((ext_vector_type(2))) float v2f;
typedef __attribute__((ext_vector_type(8))) float v8f;

__device__ __forceinline__ v8f wmma_k4(v2f a, v2f b, v8f c) {
  // V_WMMA_F32_16X16X4_F32 : D = A(16x4) * B(4x16) + C(16x16)
  return __builtin_amdgcn_wmma_f32_16x16x4_f32(
      /*neg_a=*/false, a, /*neg_b=*/false, b,
      /*c_mod=*/(short)0, c, /*reuse_a=*/false, /*reuse_b=*/false);
}

// Quadrant-correct atan2, minimax deg-11 odd poly on [0,1] (max err ~3e-7),
// v_rcp_f32 for the ratio. Matches jnp.arctan2 branch cuts via copysign.
__device__ __forceinline__ float fast_atan2f(float y, float x) {
  const float ax = __builtin_fabsf(x);
  const float ay = __builtin_fabsf(y);
  const float mx = fmaxf(ax, ay);
  const float mn = fminf(ax, ay);
  float t = (mx > 0.0f) ? mn * __builtin_amdgcn_rcpf(mx) : 0.0f; // [0,1]
  const float s = t * t;
  float p = -0.01172120f;
  p = __builtin_fmaf(p, s, 0.05265332f);
  p = __builtin_fmaf(p, s, -0.11643287f);
  p = __builtin_fmaf(p, s, 0.19354346f);
  p = __builtin_fmaf(p, s, -0.33262347f);
  p = __builtin_fmaf(p, s, 0.99997726f);
  float r = t * p;                                   // atan(mn/mx) in [0, pi/4]
  r = (ay > ax)   ? 1.57079632679f - r : r;          // octant swap
  r = (x < 0.0f)  ? 3.14159265359f - r : r;          // left half-plane
  return __builtin_copysignf(r, y);                  // sign of y (incl. +/-0 -> +/-pi)
}

// One block per 16-query tile; 8 waves split the face tiles.
__global__ __launch_bounds__(256) void winding16(
    const float* __restrict__ qpts,    // (N,3)
    const float* __restrict__ fcoord,  // (F,3,3)
    float* __restrict__ out,           // (N)
    int ftiles)
{
  __shared__ float s_acc[16];
  const int tid   = threadIdx.x;
  const int lane  = tid & 31;
  const int wave  = tid >> 5;       // 0..7
  const int half  = lane >> 4;      // 0: K={0,1}, 1: K={2,3}
  const int l16   = lane & 15;
  const int qtile = blockIdx.x;

  if (tid < 16) s_acc[tid] = 0.0f;
  __syncthreads();

  // ---- per-query-tile setup: A operand and |q|^2 C operand ----
  const float* qp = qpts + (size_t)(qtile * 16 + l16) * 3;
  const float qx = qp[0], qy = qp[1], qz = qp[2];
  v2f A;                                  // A row = [qx qy qz 1]
  A.x = half ? qz   : qx;                 // K=2 : K=0
  A.y = half ? 1.0f : qy;                 // K=3 : K=1
  const float qq_self = qx * qx + qy * qy + qz * qz;

  v8f cqq, czero = {};
  #pragma unroll
  for (int r = 0; r < 8; ++r)             // C row M = r + 8*half -> qq of that query
    cqq[r] = __shfl(qq_self, r + 8 * half, 32);

  v8f acc = {};

  for (int ft = wave; ft < ftiles; ft += 8) {
    // ---- face-side derived values (this lane owns face column l16) ----
    const float* fp = fcoord + (size_t)(ft * 16 + l16) * 9;
    const float v0x = fp[0], v0y = fp[1], v0z = fp[2];
    const float v1x = fp[3], v1y = fp[4], v1z = fp[5];
    const float v2x = fp[6], v2y = fp[7], v2z = fp[8];

    // n = v0xv1 + v1xv2 + v2xv0 ; D = v0.(v1xv2)
    const float nx = (v0y*v1z - v0z*v1y) + (v1y*v2z - v1z*v2y) + (v2y*v0z - v2z*v0y);
    const float ny = (v0z*v1x - v0x*v1z) + (v1z*v2x - v1x*v2z) + (v2z*v0x - v2x*v0z);
    const float nz = (v0x*v1y - v0y*v1x) + (v1x*v2y - v1y*v2x) + (v2x*v0y - v2y*v0x);
    const float D  = v0x*(v1y*v2z - v1z*v2y) + v0y*(v1z*v2x - v1x*v2z) + v0z*(v1x*v2y - v1y*v2x);
    const float s0  = v0x*v0x + v0y*v0y + v0z*v0z;
    const float s1  = v1x*v1x + v1y*v1y + v1z*v1z;
    const float s2  = v2x*v2x + v2y*v2y + v2z*v2z;
    const float d01 = v0x*v1x + v0y*v1y + v0z*v1z;
    const float d12 = v1x*v2x + v1y*v2y + v1z*v2z;
    const float d02 = v0x*v2x + v0y*v2y + v0z*v2z;

    // B columns: [coeff_x coeff_y coeff_z const]; lane half selects (x,y) vs (z,const)
    v2f Bna, Bnb, Bnc, Bab, Bbc, Bac, Bdet;
    Bna.x = half ? -2.0f*v0z      : -2.0f*v0x;      Bna.y = half ? s0  : -2.0f*v0y;
    Bnb.x = half ? -2.0f*v1z      : -2.0f*v1x;      Bnb.y = half ? s1  : -2.0f*v1y;
    Bnc.x = half ? -2.0f*v2z      : -2.0f*v2x;      Bnc.y = half ? s2  : -2.0f*v2y;
    Bab.x = half ? -(v0z + v1z)   : -(v0x + v1x);   Bab.y = half ? d01 : -(v0y + v1y);
    Bbc.x = half ? -(v1z + v2z)   : -(v1x + v2x);   Bbc.y = half ? d12 : -(v1y + v2y);
    Bac.x = half ? -(v0z + v2z)   : -(v0x + v2x);   Bac.y = half ? d02 : -(v0y + v2y);
    Bdet.x = half ? -nz           : -nx;            Bdet.y = half ? D  : -ny;

    // ---- 7 matrix ops: all pairwise bilinear terms for this 16x16 tile ----
    v8f na2 = wmma_k4(A, Bna, cqq);   // |a|^2
    v8f nb2 = wmma_k4(A, Bnb, cqq);   // |b|^2
    v8f nc2 = wmma_k4(A, Bnc, cqq);   // |c|^2
    v8f tab = wmma_k4(A, Bab, cqq);   // a.b
    v8f tbc = wmma_k4(A, Bbc, cqq);   // b.c
    v8f tac = wmma_k4(A, Bac, cqq);   // a.c
    v8f tdt = wmma_k4(A, Bdet, czero);// a.(b x c)

    // ---- per-pair transcendental tail (8 pairs per lane) ----
    #pragma unroll
    for (int r = 0; r < 8; ++r) {
      const float na = __builtin_amdgcn_sqrtf(fmaxf(na2[r], 0.0f));
      const float nb = __builtin_amdgcn_sqrtf(fmaxf(nb2[r], 0.0f));
      const float nc = __builtin_amdgcn_sqrtf(fmaxf(nc2[r], 0.0f));
      const float denom = na * nb * nc + tbc[r] * na + tac[r] * nb + tab[r] * nc;
      acc[r] += fast_atan2f(tdt[r], denom);
    }
  }

  // ---- reduce over the face (N) dimension: across 16 lanes of each half ----
  #pragma unroll
  for (int off = 8; off >= 1; off >>= 1) {
    #pragma unroll
    for (int r = 0; r < 8; ++r)
      acc[r] += __shfl_xor(acc[r], off, 32);
  }
  if (l16 == 0) {
    #pragma unroll
    for (int r = 0; r < 8; ++r)
      atomicAdd(&s_acc[r + 8 * half], acc[r]);   // ds_add_f32
  }
  __syncthreads();

  if (tid < 16)
    out[qtile * 16 + tid] = s_acc[tid] * 0.15915494309189535f; // 1/(2*pi)
}

extern "C" void kernel_launch(void* const* d_in, const int* in_sizes, int n_in,
                              void* d_out, int out_size, void* d_ws, size_t ws_size,
                              hipStream_t stream) {
  const float* qpts = (const float*)d_in[0];   // (1,4096,3) fp32
  const float* fc   = (const float*)d_in[1];   // (1,4096,3,3) fp32
  float* out        = (float*)d_out;           // (1,4096) fp32

  const int N = in_sizes[0] / 3;   // 4096 queries
  const int F = in_sizes[1] / 9;   // 4096 faces
  const int qtiles = N / 16;       // 256 blocks
  const int ftiles = F / 16;

  winding16<<<dim3(qtiles), dim3(256), 0, stream>>>(qpts, fc, out, ftiles);
}